// TFGWPooling_37254546326113
// MI455X (gfx1250) — compile-verified
//
#include <hip/hip_runtime.h>
#include <hip/hip_bf16.h>

// ---------------------------------------------------------------------------
// TFGW pooling for gfx1250 (MI455X).
// N=5000 nodes, E=80000 edges, F=128 feats, K=16 templates, m=10 tplt nodes.
// Dense adjacency replaced by bitmask-dedup CSR; x@F2^T and T@C2 use
// v_wmma_f32_16x16x4_f32 (clamped loads; padded-row garbage flows only to
// never-read padded outputs, so no operand masking needed except K-padding
// on B in the T@C2 kernel); Sinkhorn/linesearch as batched reduction kernels.
// ---------------------------------------------------------------------------

#define NN 5000
#define NE 80000
#define NF 128
#define NK 16
#define NM 10
#define NC (NK*NM)         // 160 columns
#define KN ((size_t)NN*NM) // 50000 per-template elements
#define W32 160            // padded words per bitmask row (ceil(5000/32)=157)
#define LOGP (-8.517193191416238f)  // log(1/5000)

typedef __attribute__((ext_vector_type(2))) float v2f;
typedef __attribute__((ext_vector_type(8))) float v8f;

// ---------------- generic helpers ----------------

__device__ __forceinline__ float blockReduceSum256(float v, float* sb) {
  int t = threadIdx.x;
  sb[t] = v; __syncthreads();
  for (int s = 128; s > 0; s >>= 1) {
    if (t < s) sb[t] += sb[t + s];
    __syncthreads();
  }
  float r = sb[0]; __syncthreads();
  return r;
}

__global__ void k_zero_u32(unsigned int* p, int n) {
  int i = blockIdx.x * 256 + threadIdx.x;
  if (i < n) p[i] = 0u;
}

// ---------------- CSR build (deduplicated adjacency) ----------------

__global__ void k_setbits(const int* __restrict__ src, const int* __restrict__ dst,
                          unsigned int* __restrict__ mask) {
  int e = blockIdx.x * 256 + threadIdx.x;
  if (e < NE) {
    int s = src[e], d = dst[e];
    atomicOr(&mask[(size_t)s * W32 + (d >> 5)], 1u << (d & 31));
  }
}

__global__ void k_rowcount(const unsigned int* __restrict__ mask,
                           int* __restrict__ rp, float* __restrict__ degp) {
  int i = blockIdx.x * 256 + threadIdx.x;
  if (i < NN) {
    int c = 0;
    for (int w = 0; w < 157; ++w) c += __popc(mask[(size_t)i * W32 + w]);
    rp[i + 1] = c;
    degp[i] = (float)c * (1.0f / (float)NN);   // (A*A)@p row, A is 0/1
    if (i == 0) rp[0] = 0;
  }
}

__global__ void k_scan(int* rp) {
  if (threadIdx.x == 0 && blockIdx.x == 0) {
    int acc = 0;
    for (int i = 1; i <= NN; ++i) { acc += rp[i]; rp[i] = acc; }
  }
}

__global__ void k_fillcsr(const unsigned int* __restrict__ mask,
                          const int* __restrict__ rp, int* __restrict__ ci) {
  int i = blockIdx.x * 256 + threadIdx.x;
  if (i < NN) {
    int pos = rp[i];
    for (int w = 0; w < 157; ++w) {
      unsigned int v = mask[(size_t)i * W32 + w];
      while (v) { int b = __ffs(v) - 1; ci[pos++] = w * 32 + b; v &= v - 1; }
    }
  }
}

// ---------------- small precompute ----------------

__global__ void k_xnorm(const float* __restrict__ x, float* __restrict__ xn) {
  int i = blockIdx.x * 256 + threadIdx.x;
  if (i < NN) {
    float s = 0.f;
    const float* xr = x + (size_t)i * NF;
#pragma unroll 4
    for (int f = 0; f < NF; f += 2) {
      v2f v = *(const v2f*)(xr + f);
      s += v.x * v.x + v.y * v.y;
    }
    xn[i] = s;
  }
}

__global__ void k_small(const float* __restrict__ F2, const float* __restrict__ C2,
                        const float* __restrict__ q0, const float* __restrict__ a0,
                        float* fn, float* cc, float* qs, float* lq, float* alphap) {
  int t = threadIdx.x;
  if (t < NC) {
    int k = t / NM, j = t - k * NM;
    const float* fr = F2 + (size_t)t * NF;
    float s = 0.f;
    for (int f = 0; f < NF; ++f) s += fr[f] * fr[f];
    fn[t] = s;
    // softmax over template-node row (each thread recomputes its row locally)
    const float* qr = q0 + k * NM;
    float mx = qr[0];
    for (int l = 1; l < NM; ++l) mx = fmaxf(mx, qr[l]);
    float se = 0.f, ql[NM];
    for (int l = 0; l < NM; ++l) { ql[l] = expf(qr[l] - mx); se += ql[l]; }
    float qj = ql[j] / se;
    qs[t] = qj;
    lq[t] = (qr[j] - mx) - logf(se);
    // (C2*C2) @ q  row j
    const float* cr = C2 + k * (NM * NM) + j * NM;
    float c = 0.f;
    for (int l = 0; l < NM; ++l) { float cv = cr[l]; c += cv * cv * (ql[l] / se); }
    cc[t] = c;
  }
  if (t == 0) alphap[0] = 1.f / (1.f + expf(-a0[0]));
}

// ---------------- WMMA GEMM: XF = x @ F2^T  [5000x128]x[128x160] ----------------
// Rows >= NN are clamped to a legal address; their (duplicate) results land
// only in padded XF rows that are never read, so no operand masking at all.
// Operands fetched in bursts of 16 b64 loads, then 8 chained WMMAs.

__global__ __launch_bounds__(32) void k_gemm_xf(const float* __restrict__ x,
                                                const float* __restrict__ F2,
                                                float* __restrict__ XF) {
  int lane = threadIdx.x;
  int tm = blockIdx.x;           // 313 row tiles (5008 padded rows)
  int tn = blockIdx.y;           // 10 col tiles (160 cols)
  int l15 = lane & 15, hi = lane >> 4;
  int row = tm * 16 + l15;
  int col = tn * 16 + l15;
  int rc = row < NN ? row : (NN - 1);
  const float* arow = x + (size_t)rc * NF;
  const float* brow = F2 + (size_t)col * NF;  // F2 flattened [160][128]
  v8f acc = {};
#pragma unroll
  for (int c = 0; c < 4; ++c) {
    v2f A[8], B[8];
#pragma unroll
    for (int s = 0; s < 8; ++s) {
      int k0 = c * 32 + s * 4 + 2 * hi;
      A[s] = *(const v2f*)(arow + k0);   // 8B aligned (k0 even)
      B[s] = *(const v2f*)(brow + k0);
    }
#pragma unroll
    for (int s = 0; s < 8; ++s)
      acc = __builtin_amdgcn_wmma_f32_16x16x4_f32(false, A[s], false, B[s],
                                                  (short)0, acc, false, false);
  }
  int mbase = hi * 8;
#pragma unroll
  for (int r = 0; r < 8; ++r)
    XF[(size_t)(tm * 16 + mbase + r) * NC + tn * 16 + l15] = acc[r];
}

// ---------------- WMMA: Out = Tsrc @ C2[k]  ([5000x10]x[10x10], padded) ----------
// K padded 10->12; only the K-padding needs masking, and zeroing the B rows
// k=10,11 suffices (kills the padded-K contribution for every output).
// Clamped A rows / extra B columns pollute only never-stored outputs.

__global__ __launch_bounds__(32) void k_tc2(const float* __restrict__ Tsrc,
                                            const float* __restrict__ C2,
                                            float* __restrict__ Out) {
  int lane = threadIdx.x;
  int tm = blockIdx.x;           // 313 row tiles
  int k  = blockIdx.y;           // template
  int l15 = lane & 15, hi = lane >> 4;
  int row = tm * 16 + l15;
  int rc = row < NN ? row : (NN - 1);
  int l9 = l15 < (NM - 1) ? l15 : (NM - 1);
  bool cv = l15 < NM;
  const float* Tr = Tsrc + (size_t)k * KN + (size_t)rc * NM;
  const float* Cb = C2 + k * (NM * NM);
  v8f acc = {};
#pragma unroll
  for (int s = 0; s < 3; ++s) {
    int k0 = s * 4 + 2 * hi;               // 0,2 | 4,6 | 8,10
    bool kv = k0 < NM;                     // only k0==10 invalid
    int kc = kv ? k0 : (NM - 2);
    v2f a = *(const v2f*)(Tr + kc);        // 8B aligned (kc even, row stride 40B)
    v2f b;
    b.x = kv ? Cb[kc * NM + l9]       : 0.f;
    b.y = kv ? Cb[(kc + 1) * NM + l9] : 0.f;
    acc = __builtin_amdgcn_wmma_f32_16x16x4_f32(false, a, false, b,
                                                (short)0, acc, false, false);
  }
  if (cv) {
    int mbase = hi * 8;
#pragma unroll
    for (int r = 0; r < 8; ++r) {
      int rr = tm * 16 + mbase + r;
      if (rr < NN) Out[(size_t)k * KN + (size_t)rr * NM + l15] = acc[r];
    }
  }
}

// ---------------- sparse A @ Y (CSR gather, deterministic) ----------------
// Rows are 40B (8B aligned): gather as 5 x b64, prefetch next neighbor row.

__global__ void k_spmm(const int* __restrict__ rp, const int* __restrict__ ci,
                       const float* __restrict__ Y, float* __restrict__ O) {
  int i = blockIdx.x * 256 + threadIdx.x;
  int k = blockIdx.y;
  if (i < NN) {
    v2f acc[5];
#pragma unroll
    for (int j = 0; j < 5; ++j) acc[j] = (v2f){0.f, 0.f};
    const float* Yb = Y + (size_t)k * KN;
    int e0 = rp[i], e1 = rp[i + 1];
    for (int e = e0; e < e1; ++e) {
      const float* yr = Yb + (size_t)ci[e] * NM;
      if (e + 1 < e1) __builtin_prefetch(Yb + (size_t)ci[e + 1] * NM, 0, 0);
#pragma unroll
      for (int j = 0; j < 5; ++j) {
        v2f v = *(const v2f*)(yr + 2 * j);
        acc[j].x += v.x; acc[j].y += v.y;
      }
    }
    float* o = O + (size_t)k * KN + (size_t)i * NM;
#pragma unroll
    for (int j = 0; j < 5; ++j) *(v2f*)(o + 2 * j) = acc[j];
  }
}

// ---------------- elementwise + reductions ----------------

__global__ void k_M(const float* __restrict__ xn, const float* __restrict__ fn,
                    const float* __restrict__ XF, float* __restrict__ M) {
  int t = blockIdx.x * 256 + threadIdx.x;
  int k = blockIdx.y;
  if (t < (int)KN) {
    int i = t / NM, j = t - i * NM;
    int c = k * NM + j;
    M[(size_t)k * KN + t] = xn[i] + fn[c] - 2.f * XF[(size_t)i * NC + c];
  }
}

__global__ void k_T0(const float* __restrict__ qs, float* __restrict__ T) {
  int t = blockIdx.x * 256 + threadIdx.x;
  int k = blockIdx.y;
  if (t < (int)KN) {
    int j = t % NM;
    T[(size_t)k * KN + t] = qs[k * NM + j] * (1.f / (float)NN);
  }
}

__global__ void k_G(const float* __restrict__ M, const float* __restrict__ ATC,
                    const float* __restrict__ degp, const float* __restrict__ cc,
                    const float* __restrict__ alphap,
                    float* __restrict__ G, float* __restrict__ absacc) {
  __shared__ float sb[256];
  int t = blockIdx.x * 256 + threadIdx.x;
  int k = blockIdx.y;
  float av = 0.f;
  if (t < (int)KN) {
    int i = t / NM, j = t - i * NM;
    float al = alphap[0];
    size_t o = (size_t)k * KN + t;
    float tens = degp[i] + cc[k * NM + j] - 2.f * ATC[o];
    float g = (1.f - al) * M[o] + 2.f * al * tens;
    G[o] = g;
    av = fabsf(g);
  }
  float s = blockReduceSum256(av, sb);
  if (threadIdx.x == 0) atomicAdd(&absacc[k], s);
}

__global__ void k_reg(const float* __restrict__ absacc, float* __restrict__ regk,
                      float* __restrict__ gv) {
  int t = threadIdx.x;
  if (t < NK) regk[t] = 0.05f * absacc[t] * (1.f / (float)KN) + 1e-9f;
  if (t < NC) gv[t] = 0.f;       // Sinkhorn g init
}

__global__ void k_sink_f(const float* __restrict__ G, const float* __restrict__ gv,
                         const float* __restrict__ regk, float* __restrict__ fv) {
  int idx = blockIdx.x * 256 + threadIdx.x;
  if (idx < NK * NN) {
    int k = idx / NN, i = idx - k * NN;
    float r = regk[k], invr = 1.f / r;
    const float* Gr = G + (size_t)k * KN + (size_t)i * NM;
    const float* gr = gv + k * NM;
    v2f vv[5];
    float mx = -3.4e38f;
#pragma unroll
    for (int p = 0; p < 5; ++p) {
      v2f Gp = *(const v2f*)(Gr + 2 * p);
      v2f gp = *(const v2f*)(gr + 2 * p);
      vv[p].x = (gp.x - Gp.x) * invr;
      vv[p].y = (gp.y - Gp.y) * invr;
      mx = fmaxf(mx, fmaxf(vv[p].x, vv[p].y));
    }
    float s = 0.f;
#pragma unroll
    for (int p = 0; p < 5; ++p) s += expf(vv[p].x - mx) + expf(vv[p].y - mx);
    fv[idx] = r * (LOGP - (mx + logf(s)));
  }
}

__global__ void k_sink_g(const float* __restrict__ G, const float* __restrict__ fv,
                         const float* __restrict__ regk, const float* __restrict__ lq,
                         float* __restrict__ gv) {
  __shared__ float sm[256], ss[256];
  int c = blockIdx.x;                  // 0..159 -> (k, j)
  int k = c / NM, j = c - k * NM;
  int t = threadIdx.x;
  float r = regk[k], invr = 1.f / r;
  const float* Gb = G + (size_t)k * KN;
  const float* fb = fv + k * NN;
  float m = -3.4e38f, s = 0.f;          // streaming logsumexp
  for (int i = t; i < NN; i += 256) {
    float v = (fb[i] - Gb[(size_t)i * NM + j]) * invr;
    if (v > m) { s = s * expf(m - v) + 1.f; m = v; }
    else        s += expf(v - m);
  }
  sm[t] = m; ss[t] = s; __syncthreads();
  for (int st = 128; st > 0; st >>= 1) {
    if (t < st) {
      float m2 = sm[t + st], s2 = ss[t + st];
      float mm = fmaxf(sm[t], m2);
      ss[t] = ss[t] * expf(sm[t] - mm) + s2 * expf(m2 - mm);
      sm[t] = mm;
    }
    __syncthreads();
  }
  if (t == 0) gv[c] = r * (lq[c] - (sm[0] + logf(ss[0])));
}

__global__ void k_dT(const float* __restrict__ G, const float* __restrict__ fv,
                     const float* __restrict__ gv, const float* __restrict__ regk,
                     const float* __restrict__ T, float* __restrict__ dT) {
  int t = blockIdx.x * 256 + threadIdx.x;
  int k = blockIdx.y;
  if (t < (int)KN) {
    int i = t / NM, j = t - i * NM;
    float r = regk[k];
    size_t o = (size_t)k * KN + t;
    float Tn = expf((fv[k * NN + i] + gv[k * NM + j] - G[o]) / r);
    dT[o] = Tn - T[o];
  }
}

__global__ void k_dots(const float* __restrict__ AdTC, const float* __restrict__ dT,
                       const float* __restrict__ M, const float* __restrict__ ATC,
                       const float* __restrict__ degp, const float* __restrict__ cc,
                       float* __restrict__ dots) {
  __shared__ float sb[256];
  int t = blockIdx.x * 256 + threadIdx.x;
  int k = blockIdx.y;
  float d0 = 0.f, d1 = 0.f, d2 = 0.f, d3 = 0.f;
  if (t < (int)KN) {
    int i = t / NM, j = t - i * NM;
    size_t o = (size_t)k * KN + t;
    float dv = dT[o];
    d0 = AdTC[o] * dv;
    d1 = M[o] * dv;
    d2 = (degp[i] + cc[k * NM + j]) * dv;
    d3 = ATC[o] * dv;
  }
  float s0 = blockReduceSum256(d0, sb);
  float s1 = blockReduceSum256(d1, sb);
  float s2 = blockReduceSum256(d2, sb);
  float s3 = blockReduceSum256(d3, sb);
  if (threadIdx.x == 0) {
    atomicAdd(&dots[k * 4 + 0], s0); atomicAdd(&dots[k * 4 + 1], s1);
    atomicAdd(&dots[k * 4 + 2], s2); atomicAdd(&dots[k * 4 + 3], s3);
  }
}

__global__ void k_gamma(const float* __restrict__ dots,
                        const float* __restrict__ alphap, float* __restrict__ gam) {
  int k = threadIdx.x;
  if (k < NK) {
    float al = alphap[0];
    float a = -2.f * al * dots[k * 4 + 0];
    float b = (1.f - al) * dots[k * 4 + 1] + al * (dots[k * 4 + 2] - 4.f * dots[k * 4 + 3]);
    float g;
    if (a > 0.f) { g = -b / (2.f * a + 1e-16f); g = fminf(fmaxf(g, 0.f), 1.f); }
    else g = (a + b < 0.f) ? 1.f : 0.f;
    gam[k] = g;
  }
}

__global__ void k_update(float* __restrict__ T, const float* __restrict__ dT,
                         const float* __restrict__ gam) {
  int t = blockIdx.x * 256 + threadIdx.x;
  int k = blockIdx.y;
  if (t < (int)KN) {
    size_t o = (size_t)k * KN + t;
    T[o] += gam[k] * dT[o];
  }
}

__global__ void k_fdots(const float* __restrict__ M, const float* __restrict__ T,
                        const float* __restrict__ ATC, const float* __restrict__ degp,
                        const float* __restrict__ cc, float* __restrict__ fd) {
  __shared__ float sb[256];
  int t = blockIdx.x * 256 + threadIdx.x;
  int k = blockIdx.y;
  float d0 = 0.f, d1 = 0.f, d2 = 0.f;
  if (t < (int)KN) {
    int i = t / NM, j = t - i * NM;
    size_t o = (size_t)k * KN + t;
    float tv = T[o];
    d0 = M[o] * tv;
    d1 = (degp[i] + cc[k * NM + j]) * tv;
    d2 = ATC[o] * tv;
  }
  float s0 = blockReduceSum256(d0, sb);
  float s1 = blockReduceSum256(d1, sb);
  float s2 = blockReduceSum256(d2, sb);
  if (threadIdx.x == 0) {
    atomicAdd(&fd[k * 3 + 0], s0); atomicAdd(&fd[k * 3 + 1], s1);
    atomicAdd(&fd[k * 3 + 2], s2);
  }
}

__global__ void k_out(const float* __restrict__ fd, const float* __restrict__ alphap,
                      float* __restrict__ out) {
  int k = threadIdx.x;
  if (k < NK) {
    float al = alphap[0];
    out[k] = (1.f - al) * fd[k * 3 + 0] + al * (fd[k * 3 + 1] - 2.f * fd[k * 3 + 2]);
  }
}

// ---------------------------------------------------------------------------

extern "C" void kernel_launch(void* const* d_in, const int* in_sizes, int n_in,
                              void* d_out, int out_size, void* d_ws, size_t ws_size,
                              hipStream_t stream) {
  const float* x  = (const float*)d_in[0];
  const int*   ei = (const int*)d_in[1];          // [2][NE]
  const float* C2 = (const float*)d_in[2];        // [16][10][10]
  const float* F2 = (const float*)d_in[3];        // [16][10][128]
  const float* q0 = (const float*)d_in[4];        // [16][10]
  const float* a0 = (const float*)d_in[5];        // [1]
  float* out = (float*)d_out;
  char* w = (char*)d_ws;

  size_t o = 0;
  auto carve = [&](size_t bytes) -> size_t {
    size_t r = o; o += (bytes + 255) & ~(size_t)255; return r;
  };
  size_t oMask = carve((size_t)NN * W32 * 4);
  size_t oRp   = carve((size_t)(NN + 1) * 4);
  size_t oCi   = carve((size_t)NE * 4);
  size_t oDeg  = carve((size_t)NN * 4);
  size_t oXn   = carve((size_t)NN * 4);
  size_t oFn   = carve(NC * 4);
  size_t oCc   = carve(NC * 4);
  size_t oQs   = carve(NC * 4);
  size_t oLq   = carve(NC * 4);
  size_t oAl   = carve(64);
  size_t oXF   = carve((size_t)5008 * NC * 4);
  size_t oM    = carve((size_t)NK * KN * 4);
  size_t oT    = carve((size_t)NK * KN * 4);
  size_t oTC2  = carve((size_t)NK * KN * 4);
  size_t oATC  = carve((size_t)NK * KN * 4);
  size_t oAdT  = carve((size_t)NK * KN * 4);
  size_t oG    = carve((size_t)NK * KN * 4);
  size_t oDT   = carve((size_t)NK * KN * 4);
  size_t oF    = carve((size_t)NK * NN * 4);
  size_t oGv   = carve(NC * 4);
  size_t oAbs  = carve(64);
  size_t oReg  = carve(64);
  size_t oDots = carve(NK * 4 * 4);
  size_t oGam  = carve(64);
  size_t oFd   = carve(NK * 3 * 4);

  unsigned int* mask = (unsigned int*)(w + oMask);
  int* rp = (int*)(w + oRp);
  int* ci = (int*)(w + oCi);
  float* degp = (float*)(w + oDeg);
  float* xn   = (float*)(w + oXn);
  float* fn   = (float*)(w + oFn);
  float* cc   = (float*)(w + oCc);
  float* qs   = (float*)(w + oQs);
  float* lq   = (float*)(w + oLq);
  float* alp  = (float*)(w + oAl);
  float* XF   = (float*)(w + oXF);
  float* Mb   = (float*)(w + oM);
  float* Tb   = (float*)(w + oT);
  float* TC2  = (float*)(w + oTC2);
  float* ATC  = (float*)(w + oATC);
  float* AdT  = (float*)(w + oAdT);
  float* Gb   = (float*)(w + oG);
  float* dT   = (float*)(w + oDT);
  float* fv   = (float*)(w + oF);
  float* gv   = (float*)(w + oGv);
  float* absacc = (float*)(w + oAbs);
  float* regk = (float*)(w + oReg);
  float* dots = (float*)(w + oDots);
  float* gam  = (float*)(w + oGam);
  float* fd   = (float*)(w + oFd);

  const int maskWords = NN * W32;
  dim3 bN((NN + 255) / 256);                 // 20 blocks over nodes
  dim3 bKN((int)((KN + 255) / 256), NK);     // 196 x 16 over [k][i*10+j]
  dim3 bTile(313, NK);                        // WMMA row tiles x templates

  // ---- precompute ----
  k_zero_u32<<<dim3((maskWords + 255) / 256), 256, 0, stream>>>(mask, maskWords);
  k_setbits<<<dim3((NE + 255) / 256), 256, 0, stream>>>(ei, ei + NE, mask);
  k_rowcount<<<bN, 256, 0, stream>>>(mask, rp, degp);
  k_scan<<<1, 1, 0, stream>>>(rp);
  k_fillcsr<<<bN, 256, 0, stream>>>(mask, rp, ci);
  k_xnorm<<<bN, 256, 0, stream>>>(x, xn);
  k_small<<<1, 256, 0, stream>>>(F2, C2, q0, a0, fn, cc, qs, lq, alp);
  k_gemm_xf<<<dim3(313, 10), 32, 0, stream>>>(x, F2, XF);
  k_M<<<bKN, 256, 0, stream>>>(xn, fn, XF, Mb);
  k_T0<<<bKN, 256, 0, stream>>>(qs, Tb);

  // ---- Frank-Wolfe outer loop (3 iterations) ----
  for (int fw = 0; fw < 3; ++fw) {
    k_tc2<<<bTile, 32, 0, stream>>>(Tb, C2, TC2);
    k_spmm<<<dim3(bN.x, NK), 256, 0, stream>>>(rp, ci, TC2, ATC);
    k_zero_u32<<<1, 256, 0, stream>>>((unsigned int*)absacc, NK);
    k_G<<<bKN, 256, 0, stream>>>(Mb, ATC, degp, cc, alp, Gb, absacc);
    k_reg<<<1, 256, 0, stream>>>(absacc, regk, gv);
    for (int it = 0; it < 10; ++it) {       // log-domain Sinkhorn
      k_sink_f<<<dim3((NK * NN + 255) / 256), 256, 0, stream>>>(Gb, gv, regk, fv);
      k_sink_g<<<dim3(NC), 256, 0, stream>>>(Gb, fv, regk, lq, gv);
    }
    k_dT<<<bKN, 256, 0, stream>>>(Gb, fv, gv, regk, Tb, dT);
    k_tc2<<<bTile, 32, 0, stream>>>(dT, C2, TC2);
    k_spmm<<<dim3(bN.x, NK), 256, 0, stream>>>(rp, ci, TC2, AdT);
    k_zero_u32<<<1, 256, 0, stream>>>((unsigned int*)dots, NK * 4);
    k_dots<<<bKN, 256, 0, stream>>>(AdT, dT, Mb, ATC, degp, cc, dots);
    k_gamma<<<1, 32, 0, stream>>>(dots, alp, gam);
    k_update<<<bKN, 256, 0, stream>>>(Tb, dT, gam);
  }

  // ---- final FGW distance per template ----
  k_tc2<<<bTile, 32, 0, stream>>>(Tb, C2, TC2);
  k_spmm<<<dim3(bN.x, NK), 256, 0, stream>>>(rp, ci, TC2, ATC);
  k_zero_u32<<<1, 256, 0, stream>>>((unsigned int*)fd, NK * 3);
  k_fdots<<<bKN, 256, 0, stream>>>(Mb, Tb, ATC, degp, cc, fd);
  k_out<<<1, 32, 0, stream>>>(fd, alp, out);

  (void)in_sizes; (void)n_in; (void)out_size; (void)ws_size;
}